// Aggregation_59433757442495
// MI455X (gfx1250) — compile-verified
//
#include <hip/hip_runtime.h>
#include <hip/hip_bf16.h>

#define D_MODEL 512
#define LQ 128
#define LM 256
#define NBATCH 8

typedef float v2f __attribute__((ext_vector_type(2)));
typedef float v8f __attribute__((ext_vector_type(8)));

__device__ __forceinline__ float fast_tanh(float x) {
#if __has_builtin(__builtin_amdgcn_tanhf)
  return __builtin_amdgcn_tanhf(x);            // v_tanh_f32 (CDNA5 TRANS pipe)
#else
  float e = __builtin_amdgcn_exp2f(x * 2.8853900817779268f); // exp(2x)
  return (e - 1.0f) * __builtin_amdgcn_rcpf(e + 1.0f);
#endif
}

__device__ __forceinline__ float fast_exp(float x) {
  return __builtin_amdgcn_exp2f(x * 1.4426950408889634f);
}

// ---------------------------------------------------------------------------
// Batched GEMM: C[bz] = A[bz] @ W[bz] (+ bias), fp32 row-major.
// N,K are template constants so all load/store offsets fold into the 24-bit
// instruction immediate (no per-iteration 64-bit address math).
// One wave computes a 64x16 strip: 4 accumulators (M-tiles) share one B frag,
// so each k-step issues 4 A-loads(b64) + 2 B-loads(b32) for 4 v_wmma.
// Grid: (N/64, M/64, batch), block: 128 (4 waves tile along N).
// Fragment layout (wave32):
//   A 16x4 : lane L holds row m=L%16, k = 2*(L/16)+i  (i = vgpr 0/1)
//   B 4x16 : lane L holds col n=L%16, k = 2*(L/16)+i
//   C 16x16: vgpr v, lane L -> row v + 8*(L/16), col L%16
// ---------------------------------------------------------------------------
template <int N, int K>
__global__ void __launch_bounds__(128)
wmma_gemm_f32(const float* __restrict__ A, const float* __restrict__ W,
              const float* __restrict__ bias, float* __restrict__ C,
              long sA, long sW, long sC) {
  const int lane = threadIdx.x & 31;
  const int wv   = threadIdx.x >> 5;
  const int hf   = lane >> 4;     // 0 or 1
  const int lr   = lane & 15;
  const int tn   = blockIdx.x * 4 + wv;   // 16-col tile
  const int tm   = blockIdx.y;            // 64-row tile
  const int bz   = blockIdx.z;

  const float* Ab = A + (size_t)bz * sA;
  const float* Wb = W + (size_t)bz * sW;
  float*       Cb = C + (size_t)bz * sC;

  const int col  = tn * 16 + lr;
  const int row0 = tm * 64 + lr;

  const float* ap0 = Ab + (size_t)(row0 +  0) * K + 2 * hf;
  const float* ap1 = Ab + (size_t)(row0 + 16) * K + 2 * hf;
  const float* ap2 = Ab + (size_t)(row0 + 32) * K + 2 * hf;
  const float* ap3 = Ab + (size_t)(row0 + 48) * K + 2 * hf;
  const float* bp  = Wb + (size_t)(2 * hf) * N + col;

  v8f c0 = {}, c1 = {}, c2 = {}, c3 = {};
  #pragma unroll 16
  for (int k = 0; k < K; k += 4) {
    v2f a0 = *(const v2f*)(ap0 + k);     // 8B aligned (k, 2*hf even)
    v2f a1 = *(const v2f*)(ap1 + k);
    v2f a2 = *(const v2f*)(ap2 + k);
    v2f a3 = *(const v2f*)(ap3 + k);
    v2f b;
    b[0] = bp[(size_t)k * N];            // row k+2*hf
    b[1] = bp[(size_t)(k + 1) * N];      // row k+2*hf+1
    c0 = __builtin_amdgcn_wmma_f32_16x16x4_f32(false, a0, false, b, (short)0, c0, false, false);
    c1 = __builtin_amdgcn_wmma_f32_16x16x4_f32(false, a1, false, b, (short)0, c1, false, false);
    c2 = __builtin_amdgcn_wmma_f32_16x16x4_f32(false, a2, false, b, (short)0, c2, false, false);
    c3 = __builtin_amdgcn_wmma_f32_16x16x4_f32(false, a3, false, b, (short)0, c3, false, false);
  }

  const float badd = bias ? bias[col] : 0.0f;
  float* cp = Cb + (size_t)(tm * 64 + 8 * hf) * N + col;
  #pragma unroll
  for (int v = 0; v < 8; ++v) {
    cp[(size_t)(v +  0) * N] = c0[v] + badd;
    cp[(size_t)(v + 16) * N] = c1[v] + badd;
    cp[(size_t)(v + 32) * N] = c2[v] + badd;
    cp[(size_t)(v + 48) * N] = c3[v] + badd;
  }
}

// ---------------------------------------------------------------------------
// Fused additive-attention scores + masked softmax.
// One 256-thread block per (b,q); thread t owns memory slot m=t.
//   S[m] = sum_d tanh(qe[b,q,d] + me[b,m,d]) * wst[d]
// qe row and wst staged in LDS, read back as float4 (ds_load_b128).
// Writes normalized probabilities P[b,q,m] to workspace.
// ---------------------------------------------------------------------------
__global__ void __launch_bounds__(256)
score_softmax(const float* __restrict__ qe, const float* __restrict__ me,
              const float* __restrict__ wst, const unsigned char* __restrict__ mask,
              float* __restrict__ P) {
  __shared__ __align__(16) float qsh[D_MODEL];
  __shared__ __align__(16) float wsh[D_MODEL];
  __shared__ float rbuf[LM];

  const int t   = threadIdx.x;
  const int bqi = blockIdx.x;           // b*LQ + q
  const int b   = bqi / LQ;

  const float* qrow = qe + (size_t)bqi * D_MODEL;
  qsh[t]       = qrow[t];
  qsh[t + 256] = qrow[t + 256];
  wsh[t]       = wst[t];
  wsh[t + 256] = wst[t + 256];
  __syncthreads();

  const float* mrow = me + ((size_t)b * LM + t) * D_MODEL;
  float s = 0.0f;
  #pragma unroll 4
  for (int d = 0; d < D_MODEL; d += 4) {
    float4 mv = *(const float4*)(mrow + d);        // global_load_b128
    float4 qv = *(const float4*)(qsh + d);         // ds_load_b128 (broadcast)
    float4 wv = *(const float4*)(wsh + d);         // ds_load_b128 (broadcast)
    s += fast_tanh(qv.x + mv.x) * wv.x;
    s += fast_tanh(qv.y + mv.y) * wv.y;
    s += fast_tanh(qv.z + mv.z) * wv.z;
    s += fast_tanh(qv.w + mv.w) * wv.w;
  }
  if (!mask[b * LM + t]) s = -__builtin_inff();

  // block max
  rbuf[t] = s;
  __syncthreads();
  #pragma unroll
  for (int off = 128; off > 0; off >>= 1) {
    if (t < off) rbuf[t] = fmaxf(rbuf[t], rbuf[t + off]);
    __syncthreads();
  }
  const float smax = rbuf[0];
  __syncthreads();

  // block sum of exp
  const float p = fast_exp(s - smax);
  rbuf[t] = p;
  __syncthreads();
  #pragma unroll
  for (int off = 128; off > 0; off >>= 1) {
    if (t < off) rbuf[t] += rbuf[t + off];
    __syncthreads();
  }
  const float inv = __builtin_amdgcn_rcpf(rbuf[0]);

  P[(size_t)bqi * LM + t] = p * inv;
}

extern "C" void kernel_launch(void* const* d_in, const int* in_sizes, int n_in,
                              void* d_out, int out_size, void* d_ws, size_t ws_size,
                              hipStream_t stream) {
  const float*         query  = (const float*)d_in[0];   // [8,128,512]
  const float*         memory = (const float*)d_in[1];   // [8,256,512]
  const float*         Wq     = (const float*)d_in[2];   // [512,512]
  const float*         bqv    = (const float*)d_in[3];   // [512]
  const float*         Wm     = (const float*)d_in[4];   // [512,512]
  const float*         bmv    = (const float*)d_in[5];   // [512]
  const float*         wst    = (const float*)d_in[6];   // [512]
  const unsigned char* mmask  = (const unsigned char*)d_in[7]; // [8,256] bool
  float* out = (float*)d_out;                             // [8,128,512]

  char* ws = (char*)d_ws;
  float* qe = (float*)(ws);                               // 8*128*512 f32 = 2MB
  float* me = (float*)(ws + (size_t)2 * 1024 * 1024);     // 8*256*512 f32 = 4MB
  float* P  = (float*)(ws + (size_t)6 * 1024 * 1024);     // 8*128*256 f32 = 1MB

  // qe = query @ Wq + bq        M=1024, N=512, K=512
  wmma_gemm_f32<512, 512><<<dim3(8, 1024 / 64, 1), 128, 0, stream>>>(
      query, Wq, bqv, qe, 0, 0, 0);

  // me = memory @ Wm + bm       M=2048, N=512, K=512
  wmma_gemm_f32<512, 512><<<dim3(8, 2048 / 64, 1), 128, 0, stream>>>(
      memory, Wm, bmv, me, 0, 0, 0);

  // P = softmax(mask(tanh-scores))   one block per (b,q)
  score_softmax<<<NBATCH * LQ, 256, 0, stream>>>(qe, me, wst, mmask, P);

  // out[b] = P[b] @ memory[b]   M=128, N=512, K=256, 8 batches
  wmma_gemm_f32<512, 256><<<dim3(8, LQ / 64, NBATCH), 128, 0, stream>>>(
      P, memory, nullptr, out,
      (long)LQ * LM, (long)LM * D_MODEL, (long)LQ * D_MODEL);
}